// ReactionTerm_56135222559103
// MI455X (gfx1250) — compile-verified
//
#include <hip/hip_runtime.h>
#include <stdint.h>

// Problem constants (from reference)
#define B_DIM   4096
#define N_SPEC  1024
#define N_R1    2048
#define N_R2    8192

#define ROWS    4      // batch rows per workgroup
#define BLOCK   256    // 8 wave32s

// ---- CDNA5 async LDS<->global data movers (gfx1250, ASYNCcnt-tracked) ----
__device__ __forceinline__ uint32_t lds_off_u32(const void* p) {
  // generic-space shared pointer: LDS byte offset lives in the low 32 bits
  return (uint32_t)(uintptr_t)p;
}

__device__ __forceinline__ void async_load_b128(uint32_t lds_off, const void* gaddr) {
  // copies 16 bytes from global address gaddr to LDS offset lds_off   (GLOBAL_LOAD_ASYNC_TO_LDS_B128)
  asm volatile("global_load_async_to_lds_b128 %0, %1, off"
               :
               : "v"(lds_off), "v"(gaddr)
               : "memory");
}

__device__ __forceinline__ void async_store_b128(const void* gaddr, uint32_t lds_off) {
  // copies 16 bytes from LDS offset lds_off to global address gaddr   (GLOBAL_STORE_ASYNC_FROM_LDS_B128)
  asm volatile("global_store_async_from_lds_b128 %0, %1, off"
               :
               : "v"(gaddr), "v"(lds_off)
               : "memory");
}

__device__ __forceinline__ void wait_async_zero() {
  asm volatile("s_wait_asynccnt 0" ::: "memory");
}

__global__ __launch_bounds__(BLOCK) void ReactionTerm_kernel(
    const float* __restrict__ t_in,   // [B]
    const float* __restrict__ y_in,   // [B, N_SPEC]
    const float* __restrict__ k1,     // [N_R1]
    const float* __restrict__ k2,     // [N_R2]
    const int*   __restrict__ i1r,    // [N_R1]
    const int*   __restrict__ i1p,    // [N_R1]
    const int*   __restrict__ i2r,    // [N_R2, 2] flat
    const int*   __restrict__ i2p,    // [N_R2]
    float*       __restrict__ y_out)  // [B, N_SPEC]
{
  __shared__ float y_s[ROWS * N_SPEC];    // 16 KB staged input rows
  __shared__ float acc_s[ROWS * N_SPEC];  // 16 KB output accumulators

  const int tid = threadIdx.x;
  const int rb  = blockIdx.x * ROWS;      // first batch row of this WG
  const float* ysrc = y_in  + (size_t)rb * N_SPEC;
  float*       ydst = y_out + (size_t)rb * N_SPEC;

  // ---- Phase 1: async DMA of ROWS y-rows into LDS (1024 x b128) ----
  const uint32_t ybase = lds_off_u32(&y_s[0]);
#pragma unroll
  for (int i = 0; i < (ROWS * N_SPEC) / 4 / BLOCK; ++i) {   // 4 iters
    const int e = (i * BLOCK + tid) * 4;                    // float index
    async_load_b128(ybase + (uint32_t)e * 4u, ysrc + e);
  }

  // Zero accumulators while the async copies are in flight
#pragma unroll
  for (int i = 0; i < (ROWS * N_SPEC) / BLOCK; ++i)         // 16 iters
    acc_s[i * BLOCK + tid] = 0.0f;

  // Per-row time values (wave-uniform broadcast loads, L2-resident)
  float t[ROWS];
#pragma unroll
  for (int w = 0; w < ROWS; ++w) t[w] = t_in[rb + w];

  wait_async_zero();
  __syncthreads();

  // ---- Phase 2a: first-order reactions, 4 per thread via b128 packets ----
  // N_R1 = 2048 = 2 sweeps of (BLOCK*4)
  for (int base = 0; base < N_R1; base += BLOCK * 4) {
    const int r0 = base + tid * 4;
    const int4   ir4 = *(const int4*)  (i1r + r0);
    const int4   ip4 = *(const int4*)  (i1p + r0);
    const float4 k4  = *(const float4*)(k1  + r0);
    const int   irs[4] = {ir4.x, ir4.y, ir4.z, ir4.w};
    const int   ips[4] = {ip4.x, ip4.y, ip4.z, ip4.w};
    const float ks [4] = {k4.x,  k4.y,  k4.z,  k4.w};
#pragma unroll
    for (int j = 0; j < 4; ++j) {
#pragma unroll
      for (int w = 0; w < ROWS; ++w) {
        const float term = y_s[w * N_SPEC + irs[j]] * (t[w] * ks[j]);
        atomicAdd(&acc_s[w * N_SPEC + ips[j]], term);       // ds_add_f32
      }
    }
  }

  // ---- Phase 2b: second-order reactions, 4 per thread via b128 packets ----
  // N_R2 = 8192 = 8 sweeps of (BLOCK*4)
  for (int base = 0; base < N_R2; base += BLOCK * 4) {
    const int r0 = base + tid * 4;
    const int4   ra  = *(const int4*)  (i2r + 2 * r0);      // pairs for j=0,1
    const int4   rc  = *(const int4*)  (i2r + 2 * r0 + 4);  // pairs for j=2,3
    const int4   ip4 = *(const int4*)  (i2p + r0);
    const float4 k4  = *(const float4*)(k2  + r0);
    const int   i0s[4] = {ra.x, ra.z, rc.x, rc.z};
    const int   i1s[4] = {ra.y, ra.w, rc.y, rc.w};
    const int   ips[4] = {ip4.x, ip4.y, ip4.z, ip4.w};
    const float ks [4] = {k4.x,  k4.y,  k4.z,  k4.w};
#pragma unroll
    for (int j = 0; j < 4; ++j) {
#pragma unroll
      for (int w = 0; w < ROWS; ++w) {
        const float term = y_s[w * N_SPEC + i0s[j]] *
                           y_s[w * N_SPEC + i1s[j]] * (t[w] * ks[j]);
        atomicAdd(&acc_s[w * N_SPEC + ips[j]], term);       // ds_add_f32
      }
    }
  }

  __syncthreads();

  // ---- Phase 3: async DMA of accumulators LDS -> global (coalesced b128) ----
  const uint32_t abase = lds_off_u32(&acc_s[0]);
#pragma unroll
  for (int i = 0; i < (ROWS * N_SPEC) / 4 / BLOCK; ++i) {   // 4 iters
    const int e = (i * BLOCK + tid) * 4;
    async_store_b128(ydst + e, abase + (uint32_t)e * 4u);
  }
  // S_ENDPGM performs an implicit wait-idle (all counters, incl. ASYNCcnt)
}

extern "C" void kernel_launch(void* const* d_in, const int* in_sizes, int n_in,
                              void* d_out, int out_size, void* d_ws, size_t ws_size,
                              hipStream_t stream) {
  const float* t_in = (const float*)d_in[0];
  const float* y_in = (const float*)d_in[1];
  const float* k1   = (const float*)d_in[2];
  const float* k2   = (const float*)d_in[3];
  const int*   i1r  = (const int*)  d_in[4];
  const int*   i1p  = (const int*)  d_in[5];
  const int*   i2r  = (const int*)  d_in[6];   // [N_R2,2] flat
  const int*   i2p  = (const int*)  d_in[7];
  float*       yout = (float*)d_out;

  (void)in_sizes; (void)n_in; (void)out_size; (void)d_ws; (void)ws_size;

  dim3 grid(B_DIM / ROWS);   // 1024 workgroups
  dim3 block(BLOCK);         // 8 wave32s
  ReactionTerm_kernel<<<grid, block, 0, stream>>>(
      t_in, y_in, k1, k2, i1r, i1p, i2r, i2p, yout);
}